// HyMBA_Block_46007689674771
// MI455X (gfx1250) — compile-verified
//
#include <hip/hip_runtime.h>
#include <math.h>

#define DIM  1024
#define SSM  64
#define RNN  128
#define BATCH 8
#define SEQ  2048
#define MTOT (BATCH*SEQ)

typedef float v2f __attribute__((ext_vector_type(2)));
typedef float v8f __attribute__((ext_vector_type(8)));

// Full-precision CDNA5 matrix op: D(16x16 f32) = A(16x4 f32) x B(4x16 f32) + C
__device__ __forceinline__ v8f wmma_f32(v2f a, v2f b, v8f c) {
  return __builtin_amdgcn_wmma_f32_16x16x4_f32(false, a, false, b, (short)0, c,
                                               false, false);
}

__device__ __forceinline__ float sigf(float x) { return 1.f / (1.f + expf(-x)); }

// ---------------------------------------------------------------------------
// Phase 1: Y[M,N] = X[M,K] @ W[N,K]^T + bias[N]   (fp32, WMMA 16x16x4)
// block = 256 (8 waves). Macro tile 128(M) x 64(N), per-wave 32x32 (2x2 tiles).
// Register-staged pipeline: chunk k+1 global loads overlap chunk k WMMAs;
// global_prefetch_b8 runs one further chunk ahead.
// ---------------------------------------------------------------------------
__global__ __launch_bounds__(256) void gemm_xwT(const float* __restrict__ X,
                                                const float* __restrict__ W,
                                                const float* __restrict__ bias,
                                                float* __restrict__ Y,
                                                int N, int K) {
  __shared__ float Xs[128 * 33];   // +1 padding -> conflict-free frag reads
  __shared__ float Ws[64 * 33];
  const int tid  = threadIdx.x;
  const int lane = tid & 31, wid = tid >> 5;
  const int lh = lane >> 4, ln = lane & 15;   // lane-half / lane-in-16
  const int wm = wid >> 1, wn = wid & 1;
  const int gm0 = blockIdx.x * 128;
  const int gn0 = blockIdx.y * 64;
  const float* Xg = X + (size_t)gm0 * K;
  const float* Wg = W + (size_t)gn0 * K;

  v8f acc[2][2];
#pragma unroll
  for (int a = 0; a < 2; ++a)
#pragma unroll
    for (int b = 0; b < 2; ++b)
#pragma unroll
      for (int r = 0; r < 8; ++r) acc[a][b][r] = 0.f;

  float xr[16], wr[8];                        // staging registers
#pragma unroll
  for (int i = 0; i < 16; ++i) {              // chunk 0: 128x32 X tile
    const int idx = tid + i * 256;
    xr[i] = Xg[(size_t)(idx >> 5) * K + (idx & 31)];
  }
#pragma unroll
  for (int i = 0; i < 8; ++i) {               // chunk 0: 64x32 W tile
    const int idx = tid + i * 256;
    wr[i] = Wg[(size_t)(idx >> 5) * K + (idx & 31)];
  }

  for (int k0 = 0; k0 < K; k0 += 32) {
    // commit staged registers to LDS
#pragma unroll
    for (int i = 0; i < 16; ++i) {
      const int idx = tid + i * 256;
      Xs[(idx >> 5) * 33 + (idx & 31)] = xr[i];
    }
#pragma unroll
    for (int i = 0; i < 8; ++i) {
      const int idx = tid + i * 256;
      Ws[(idx >> 5) * 33 + (idx & 31)] = wr[i];
    }
    __syncthreads();

    const int kn = k0 + 32;
    if (kn < K) {   // issue next-chunk loads; latency hides under the WMMAs
#pragma unroll
      for (int i = 0; i < 16; ++i) {
        const int idx = tid + i * 256;
        xr[i] = Xg[(size_t)(idx >> 5) * K + kn + (idx & 31)];
      }
#pragma unroll
      for (int i = 0; i < 8; ++i) {
        const int idx = tid + i * 256;
        wr[i] = Wg[(size_t)(idx >> 5) * K + kn + (idx & 31)];
      }
      if (kn + 32 < K) {   // prefetch chunk k+2 into cache (global_prefetch_b8)
        if (tid < 128)
          __builtin_prefetch(&Xg[(size_t)tid * K + kn + 32], 0, 0);
        else if (tid < 192)
          __builtin_prefetch(&Wg[(size_t)(tid - 128) * K + kn + 32], 0, 0);
      }
    }

#pragma unroll
    for (int kk = 0; kk < 32; kk += 4) {
      const int kb = kk + lh * 2;           // A frag: lanes 0-15 K=0,1; 16-31 K=2,3
      const int ar0 = wm * 32 + ln, ar1 = ar0 + 16;
      const int bn0 = wn * 32 + ln, bn1 = bn0 + 16;
      v2f a0, a1, b0, b1;
      a0.x = Xs[ar0 * 33 + kb]; a0.y = Xs[ar0 * 33 + kb + 1];
      a1.x = Xs[ar1 * 33 + kb]; a1.y = Xs[ar1 * 33 + kb + 1];
      b0.x = Ws[bn0 * 33 + kb]; b0.y = Ws[bn0 * 33 + kb + 1];
      b1.x = Ws[bn1 * 33 + kb]; b1.y = Ws[bn1 * 33 + kb + 1];
      acc[0][0] = wmma_f32(a0, b0, acc[0][0]);
      acc[0][1] = wmma_f32(a0, b1, acc[0][1]);
      acc[1][0] = wmma_f32(a1, b0, acc[1][0]);
      acc[1][1] = wmma_f32(a1, b1, acc[1][1]);
    }
    __syncthreads();
  }
#pragma unroll
  for (int mt = 0; mt < 2; ++mt)
#pragma unroll
    for (int nt = 0; nt < 2; ++nt) {
      const int n = gn0 + wn * 32 + nt * 16 + ln;
      const float bv = bias[n];
#pragma unroll
      for (int r = 0; r < 8; ++r) {
        const int m = gm0 + wm * 32 + mt * 16 + lh * 8 + r;
        Y[(size_t)m * N + n] = acc[mt][nt][r] + bv;
      }
    }
}

// ---------------------------------------------------------------------------
// Gate projection: gx[row,g] = sigmoid(x[row,:] . gate_w[g,:] + gate_b[g])
// ---------------------------------------------------------------------------
__global__ __launch_bounds__(256) void gate_proj(const float* __restrict__ X,
                                                 const float* __restrict__ gw,
                                                 const float* __restrict__ gb,
                                                 float* __restrict__ gx) {
  const int lane = threadIdx.x & 31, wid = threadIdx.x >> 5;
  const int row = blockIdx.x * 4 + (wid >> 1);
  const int g = wid & 1;
  const float* xr = X + (size_t)row * DIM;
  const float* wr = gw + (size_t)g * DIM;
  float s = 0.f;
  for (int i = lane; i < DIM; i += 32) s += xr[i] * wr[i];
#pragma unroll
  for (int m = 16; m; m >>= 1) s += __shfl_xor(s, m, 32);
  if (lane == 0) gx[(size_t)row * 2 + g] = sigf(s + gb[g]);
}

// ---------------------------------------------------------------------------
// Phase 2: sequential scan. 16 independent WGs; each replicates the tiny
// recurrence and owns a 64-channel slice of the D=1024 output projections.
// All weights + ping-pong state live in LDS (320,256 B <= 320 KB / WGP).
// Step t prefetches step t+1's L2-resident streams (bx/dx/ih/gx).
// ---------------------------------------------------------------------------
#define OFF_A    0            // A_s      [64][65]
#define OFF_WHH  4160         // Whh_s    [384][129]
#define OFF_CW   53696        // Cw_s     [64][65]   (slice)
#define OFF_PW   57856        // Pw_s     [64][129]  (slice)
#define OFF_HS   66112        // h_ssm    [2][16][65]
#define OFF_HR   68192        // h_rnn    [2][16][129]
#define OFF_HH   72320        // hh       [16][385]
#define OFF_YR   78480        // y_rnn    [16][65]
#define OFF_G    79520        // gates    [16][2]
#define OFF_BHH  79552        // b_hh     [384]
#define OFF_CB   79936        // Cb slice [64]
#define OFF_PB   80000        // proj_b   [64]
#define SCAN_SMEM_FLOATS 80064

__global__ __launch_bounds__(256) void hymba_scan(
    const float* __restrict__ bx, const float* __restrict__ dxp,
    const float* __restrict__ gx, const float* __restrict__ ihp,
    const float* __restrict__ A,  const float* __restrict__ Cw,
    const float* __restrict__ Cb, const float* __restrict__ W_hh,
    const float* __restrict__ b_hh, const float* __restrict__ pw,
    const float* __restrict__ pb, float* __restrict__ out) {
  extern __shared__ float sm[];
  float* A_s   = sm + OFF_A;
  float* Whh_s = sm + OFF_WHH;
  float* Cw_s  = sm + OFF_CW;
  float* Pw_s  = sm + OFF_PW;
  float* Hs_s  = sm + OFF_HS;
  float* Hr_s  = sm + OFF_HR;
  float* hh_s  = sm + OFF_HH;
  float* yr_s  = sm + OFF_YR;
  float* g_s   = sm + OFF_G;
  float* bhh_s = sm + OFF_BHH;
  float* Cb_s  = sm + OFF_CB;
  float* Pb_s  = sm + OFF_PB;

  const int tid = threadIdx.x, lane = tid & 31, wid = tid >> 5;
  const int lh = lane >> 4, ln = lane & 15;
  const int cbase = blockIdx.x * 64;        // this WG's output-channel slice

  // ---- one-time weight staging into LDS (amortized over 2048 steps) ----
  for (int i = tid; i < SSM * SSM; i += 256)
    A_s[(i >> 6) * 65 + (i & 63)] = A[i];
  for (int i = tid; i < 3 * RNN * RNN; i += 256)
    Whh_s[(i >> 7) * 129 + (i & 127)] = W_hh[i];
  for (int i = tid; i < 64 * SSM; i += 256)
    Cw_s[(i >> 6) * 65 + (i & 63)] = Cw[(size_t)(cbase + (i >> 6)) * SSM + (i & 63)];
  for (int i = tid; i < 64 * RNN; i += 256)
    Pw_s[(i >> 7) * 129 + (i & 127)] = pw[(size_t)(cbase + (i >> 7)) * RNN + (i & 127)];
  for (int i = tid; i < 3 * RNN; i += 256) bhh_s[i] = b_hh[i];
  if (tid < 64) { Cb_s[tid] = Cb[cbase + tid]; Pb_s[tid] = pb[cbase + tid]; }
  for (int i = tid; i < 2 * 16 * 65; i += 256) Hs_s[i] = 0.f;
  for (int i = tid; i < 2 * 16 * 129; i += 256) Hr_s[i] = 0.f;
  __syncthreads();

  for (int t = 0; t < SEQ; ++t) {
    const int cur = t & 1, nxt = cur ^ 1;
    float* hs_c = Hs_s + cur * 16 * 65;
    float* hs_n = Hs_s + nxt * 16 * 65;
    float* hr_c = Hr_s + cur * 16 * 129;
    float* hr_n = Hr_s + nxt * 16 * 129;

    // prefetch step t+1 streams (one 128B line per thread, 136 lines total)
    if (t + 1 < SEQ) {
      const int tp = t + 1;
      if (tid < 96) {                       // ih: 8 rows x 12 lines
        const int m = tid / 12, l = tid - 12 * m;
        __builtin_prefetch(&ihp[((size_t)m * SEQ + tp) * (3 * RNN) + l * 32], 0, 0);
      } else if (tid < 112) {               // dx slice: 8 rows x 2 lines
        const int q = tid - 96, m = q >> 1, l = q & 1;
        __builtin_prefetch(&dxp[((size_t)m * SEQ + tp) * DIM + cbase + l * 32], 0, 0);
      } else if (tid < 128) {               // bx: 8 rows x 2 lines
        const int q = tid - 112, m = q >> 1, l = q & 1;
        __builtin_prefetch(&bx[((size_t)m * SEQ + tp) * SSM + l * 32], 0, 0);
      } else if (tid < 136) {               // gx: 8 rows
        const int m = tid - 128;
        __builtin_prefetch(&gx[((size_t)m * SEQ + tp) * 2], 0, 0);
      }
    }

    // (a) hh = h_rnn @ W_hh^T + b_hh : M=16, N=384 (24 tiles / 8 waves), K=128
    v8f hacc[3];
#pragma unroll
    for (int j = 0; j < 3; ++j) {
      const float bv = bhh_s[(wid * 3 + j) * 16 + ln];
#pragma unroll
      for (int r = 0; r < 8; ++r) hacc[j][r] = bv;
    }
    for (int kk = 0; kk < RNN; kk += 4) {
      const int kb = kk + lh * 2;
      v2f a; a.x = hr_c[ln * 129 + kb]; a.y = hr_c[ln * 129 + kb + 1];
#pragma unroll
      for (int j = 0; j < 3; ++j) {
        const int n0 = (wid * 3 + j) * 16;
        v2f b; b.x = Whh_s[(n0 + ln) * 129 + kb];
        b.y = Whh_s[(n0 + ln) * 129 + kb + 1];
        hacc[j] = wmma_f32(a, b, hacc[j]);
      }
    }
    // waves 0-3: h_ssm' = silu(h_ssm @ A^T + bx_t) : N=64 (tile per wave), K=64
    if (wid < 4) {
      const int n0 = wid * 16;
      v8f sacc;
#pragma unroll
      for (int r = 0; r < 8; ++r) {
        const int m = lh * 8 + r;
        sacc[r] = (m < BATCH) ? bx[((size_t)m * SEQ + t) * SSM + n0 + ln] : 0.f;
      }
      for (int kk = 0; kk < SSM; kk += 4) {
        const int kb = kk + lh * 2;
        v2f a; a.x = hs_c[ln * 65 + kb]; a.y = hs_c[ln * 65 + kb + 1];
        v2f b; b.x = A_s[(n0 + ln) * 65 + kb]; b.y = A_s[(n0 + ln) * 65 + kb + 1];
        sacc = wmma_f32(a, b, sacc);
      }
#pragma unroll
      for (int r = 0; r < 8; ++r) {          // silu, pad rows stay 0
        const float v = sacc[r];
        hs_n[(lh * 8 + r) * 65 + n0 + ln] = v * sigf(v);
      }
    }
    // (b) park hh fragments
#pragma unroll
    for (int j = 0; j < 3; ++j) {
      const int n0 = (wid * 3 + j) * 16;
#pragma unroll
      for (int r = 0; r < 8; ++r)
        hh_s[(lh * 8 + r) * 385 + n0 + ln] = hacc[j][r];
    }
    __syncthreads();

    // (c) GRU pointwise (torch semantics); pad rows go nonzero (b_hh) but never reach out
    for (int idx = tid; idx < 16 * RNN; idx += 256) {
      const int m = idx >> 7, j = idx & 127;
      float ir = 0.f, iz = 0.f, inn = 0.f;
      if (m < BATCH) {
        const size_t ib = ((size_t)m * SEQ + t) * (3 * RNN) + j;
        ir = ihp[ib]; iz = ihp[ib + RNN]; inn = ihp[ib + 2 * RNN];
      }
      const float h_r = hh_s[m * 385 + j];
      const float h_z = hh_s[m * 385 + j + RNN];
      const float h_n = hh_s[m * 385 + j + 2 * RNN];
      const float hp  = hr_c[m * 129 + j];
      const float rg = sigf(ir + h_r);
      const float zg = sigf(iz + h_z);
      const float ng = tanhf(inn + rg * h_n);
      hr_n[m * 129 + j] = (1.f - zg) * ng + zg * hp;
    }
    if (tid < 32) {
      const int m = tid >> 1, g = tid & 1;
      g_s[tid] = (m < BATCH) ? gx[((size_t)m * SEQ + t) * 2 + g] : 0.f;
    }
    __syncthreads();

    // (d) output projections on this WG's 64-channel slice
    v8f cacc;
#pragma unroll
    for (int r = 0; r < 8; ++r) cacc[r] = 0.f;
    if (wid < 4) {   // y_ssm = h_ssm' @ Cw_slice^T + Cb + dx
      const int n0 = wid * 16;
#pragma unroll
      for (int r = 0; r < 8; ++r) {
        const int m = lh * 8 + r;
        float v = Cb_s[n0 + ln];
        if (m < BATCH) v += dxp[((size_t)m * SEQ + t) * DIM + cbase + n0 + ln];
        cacc[r] = v;
      }
      for (int kk = 0; kk < SSM; kk += 4) {
        const int kb = kk + lh * 2;
        v2f a; a.x = hs_n[ln * 65 + kb]; a.y = hs_n[ln * 65 + kb + 1];
        v2f b; b.x = Cw_s[(n0 + ln) * 65 + kb]; b.y = Cw_s[(n0 + ln) * 65 + kb + 1];
        cacc = wmma_f32(a, b, cacc);
      }
    } else {         // y_rnn = h_rnn' @ proj_slice^T + pb  -> stage to LDS
      const int n0 = (wid - 4) * 16;
      v8f racc;
      const float pv = Pb_s[n0 + ln];
#pragma unroll
      for (int r = 0; r < 8; ++r) racc[r] = pv;
      for (int kk = 0; kk < RNN; kk += 4) {
        const int kb = kk + lh * 2;
        v2f a; a.x = hr_n[ln * 129 + kb]; a.y = hr_n[ln * 129 + kb + 1];
        v2f b; b.x = Pw_s[(n0 + ln) * 129 + kb];
        b.y = Pw_s[(n0 + ln) * 129 + kb + 1];
        racc = wmma_f32(a, b, racc);
      }
#pragma unroll
      for (int r = 0; r < 8; ++r)
        yr_s[(lh * 8 + r) * 65 + n0 + ln] = racc[r];
    }
    __syncthreads();

    // (e) gated fusion + store (coalesced across lanes)
    if (wid < 4) {
      const int n0 = wid * 16;
#pragma unroll
      for (int r = 0; r < 8; ++r) {
        const int m = lh * 8 + r;
        if (m < BATCH) {
          const int n = n0 + ln;
          const float v = g_s[m * 2] * cacc[r] + g_s[m * 2 + 1] * yr_s[m * 65 + n];
          out[((size_t)m * SEQ + t) * DIM + cbase + n] = v;
        }
      }
    }
    // next-step writes are ordered behind this step's reads by barriers (c)/(d)
  }
}

// ---------------------------------------------------------------------------
extern "C" void kernel_launch(void* const* d_in, const int* in_sizes, int n_in,
                              void* d_out, int out_size, void* d_ws,
                              size_t ws_size, hipStream_t stream) {
  (void)in_sizes; (void)n_in; (void)out_size; (void)ws_size;
  const float* x    = (const float*)d_in[0];
  const float* A    = (const float*)d_in[1];
  const float* Bw   = (const float*)d_in[2];
  const float* Bb   = (const float*)d_in[3];
  const float* Cw   = (const float*)d_in[4];
  const float* Cb   = (const float*)d_in[5];
  const float* Dw   = (const float*)d_in[6];
  const float* Db   = (const float*)d_in[7];
  const float* W_ih = (const float*)d_in[8];
  const float* b_ih = (const float*)d_in[9];
  const float* W_hh = (const float*)d_in[10];
  const float* b_hh = (const float*)d_in[11];
  const float* pw   = (const float*)d_in[12];
  const float* pb   = (const float*)d_in[13];
  const float* gw   = (const float*)d_in[14];
  const float* gb   = (const float*)d_in[15];
  float* out = (float*)d_out;

  float* ws  = (float*)d_ws;                       // ~96.6 MB scratch
  float* bxp = ws;                                 // [MTOT, 64]
  float* dxp = bxp + (size_t)MTOT * SSM;           // [MTOT, 1024]
  float* ihp = dxp + (size_t)MTOT * DIM;           // [MTOT, 384]
  float* gxp = ihp + (size_t)MTOT * 3 * RNN;       // [MTOT, 2]

  dim3 blk(256);
  gemm_xwT<<<dim3(MTOT / 128, SSM / 64), blk, 0, stream>>>(x, Bw, Bb, bxp, SSM, DIM);
  gemm_xwT<<<dim3(MTOT / 128, DIM / 64), blk, 0, stream>>>(x, Dw, Db, dxp, DIM, DIM);
  gemm_xwT<<<dim3(MTOT / 128, (3 * RNN) / 64), blk, 0, stream>>>(x, W_ih, b_ih, ihp, 3 * RNN, DIM);
  gate_proj<<<dim3(MTOT / 4), blk, 0, stream>>>(x, gw, gb, gxp);

  const size_t smem = (size_t)SCAN_SMEM_FLOATS * sizeof(float);
  hipFuncSetAttribute(reinterpret_cast<const void*>(hymba_scan),
                      hipFuncAttributeMaxDynamicSharedMemorySize, (int)smem);
  hymba_scan<<<dim3(DIM / 64), blk, smem, stream>>>(bxp, dxp, gxp, ihp, A, Cw, Cb,
                                                    W_hh, b_hh, pw, pb, out);
}